// GCN_85040352461485
// MI455X (gfx1250) — compile-verified
//
#include <hip/hip_runtime.h>
#include <math.h>
#include <stdint.h>

#define N_NODES   20000
#define N_EDGES   640000
#define D_IN      128
#define D_HID     256
#define D_DENSE   128
#define N_CLASSES 2

typedef __attribute__((ext_vector_type(2))) float v2f;
typedef __attribute__((ext_vector_type(8))) float v8f;

__device__ __forceinline__ void atomicAddF32(float* p, float v) {
    // relaxed agent-scope f32 atomic -> global_atomic_add_f32 on gfx1250
    __hip_atomic_fetch_add(p, v, __ATOMIC_RELAXED, __HIP_MEMORY_SCOPE_AGENT);
}

// ---------------------------------------------------------------------------
// init: deg <- 1.0 (self loop weight), gsum <- 0
// ---------------------------------------------------------------------------
__global__ __launch_bounds__(256) void k_init(float* __restrict__ deg,
                                              float* __restrict__ gsum) {
    int i = blockIdx.x * 256 + threadIdx.x;
    if (i < N_NODES) deg[i] = 1.0f;
    if (i < D_HID)   gsum[i] = 0.0f;
}

// deg[col[e]] += w[e]
__global__ __launch_bounds__(256) void k_deg(const int* __restrict__ col,
                                             const float* __restrict__ w,
                                             float* __restrict__ deg) {
    int e = blockIdx.x * 256 + threadIdx.x;
    if (e < N_EDGES) atomicAddF32(&deg[col[e]], w[e]);
}

// dinv = deg > 0 ? rsqrt(deg) : 0   (in place)
__global__ __launch_bounds__(256) void k_rsqrt(float* __restrict__ deg) {
    int i = blockIdx.x * 256 + threadIdx.x;
    if (i < N_NODES) {
        float d = deg[i];
        deg[i] = (d > 0.0f) ? rsqrtf(d) : 0.0f;
    }
}

// ---------------------------------------------------------------------------
// H[M x 256] = X[M x K] @ W[K x 256]   via V_WMMA_F32_16X16X4_F32
// block = 256 threads = 8 waves; block tile 32 rows x 256 cols (full N)
// wave tile 16 x 64 (four 16x16 accumulators sharing one A fragment)
// A tile staged to LDS with CDNA5 async-to-LDS (ASYNCcnt) on gfx1250
// grid = (M/32)
// ---------------------------------------------------------------------------
__global__ __launch_bounds__(256) void k_gemm(const float* __restrict__ X,
                                              const float* __restrict__ W,
                                              float* __restrict__ H,
                                              int K) {
    const int NOUT   = D_HID;             // 256
    const int m_base = blockIdx.x * 32;
    const int tid    = threadIdx.x;
    const int stride = K + 4;             // LDS row pad: bank stride 4, rows 16B aligned

    __shared__ float lds[32 * (D_HID + 4)];   // sized for K=256 (33,280 B)

    // ---- stage A tile (32 x K) into LDS ----
    const int k4   = K >> 2;
    const int nvec = 32 * k4;                 // float4 elements
#if defined(__gfx1250__)
    // async copy global -> LDS, 16B per lane per issue, tracked by ASYNCcnt.
    // VDST VGPR = LDS byte address. A generic LDS address is
    // {shared-aperture | offset} with addr[31:0] == LDS byte offset, so the
    // truncated ptrtoint of the REAL &lds[...] pointer is both the correct
    // operand value and (crucially) an escape of `lds`, so the "memory"
    // clobber forces the compiler to treat LDS as written by the asm.
    for (int idx = tid; idx < nvec; idx += 256) {
        int r  = idx / k4;
        int c4 = idx - r * k4;
        const float* gp = X + (size_t)(m_base + r) * K + (c4 << 2);
        float*       lp = &lds[r * stride + (c4 << 2)];
        unsigned loff = (unsigned)(uintptr_t)lp;         // LDS byte address
        unsigned long long ga = (unsigned long long)(uintptr_t)gp;
        asm volatile("global_load_async_to_lds_b128 %0, %1, off"
                     :: "v"(loff), "v"(ga) : "memory");
    }
    asm volatile("s_wait_asynccnt 0x0" ::: "memory");
#else
    for (int idx = tid; idx < nvec; idx += 256) {
        int r  = idx / k4;
        int c4 = idx - r * k4;
        const float4 v = *reinterpret_cast<const float4*>(
            X + (size_t)(m_base + r) * K + (c4 << 2));
        *reinterpret_cast<float4*>(&lds[r * stride + (c4 << 2)]) = v;
    }
#endif
    __syncthreads();

    const int wid   = tid >> 5;
    const int lane  = tid & 31;
    const int wm    = wid >> 2;                   // 0..1  (row tile within block)
    const int wn    = wid & 3;                    // 0..3  (col group within block)
    const int lrow  = (wm << 4) + (lane & 15);    // A-frag row in LDS tile
    const int khalf = (lane >> 4) << 1;           // lanes 0-15: K+0/1, 16-31: K+2/3
    const int n0    = (wn << 6) + (lane & 15);    // wave covers cols [wn*64, wn*64+64)

    v8f acc0 = {};
    v8f acc1 = {};
    v8f acc2 = {};
    v8f acc3 = {};
    const float* ldsrow = &lds[lrow * stride];

    #pragma unroll 4
    for (int k0 = 0; k0 < K; k0 += 4) {
        const int k = k0 + khalf;
        v2f a;
        a.x = ldsrow[k];
        a.y = ldsrow[k + 1];

        const float* wp = W + (size_t)k * NOUT + n0;
        v2f b0, b1, b2, b3;
        b0.x = wp[0];          b0.y = wp[NOUT];
        b1.x = wp[16];         b1.y = wp[NOUT + 16];
        b2.x = wp[32];         b2.y = wp[NOUT + 32];
        b3.x = wp[48];         b3.y = wp[NOUT + 48];

        acc0 = __builtin_amdgcn_wmma_f32_16x16x4_f32(false, a, false, b0,
                                                     (short)0, acc0, false, false);
        acc1 = __builtin_amdgcn_wmma_f32_16x16x4_f32(false, a, false, b1,
                                                     (short)0, acc1, false, false);
        acc2 = __builtin_amdgcn_wmma_f32_16x16x4_f32(false, a, false, b2,
                                                     (short)0, acc2, false, false);
        acc3 = __builtin_amdgcn_wmma_f32_16x16x4_f32(false, a, false, b3,
                                                     (short)0, acc3, false, false);
    }

    // C/D layout: VGPR i: lanes 0-15 -> row i, lanes 16-31 -> row i+8
    const int mrow = m_base + (wm << 4) + ((lane >> 4) << 3);
    float* Hp = H + (size_t)mrow * NOUT + n0;
    #pragma unroll
    for (int i = 0; i < 8; ++i) {
        Hp[(size_t)i * NOUT]      = acc0[i];
        Hp[(size_t)i * NOUT + 16] = acc1[i];
        Hp[(size_t)i * NOUT + 32] = acc2[i];
        Hp[(size_t)i * NOUT + 48] = acc3[i];
    }
}

// ---------------------------------------------------------------------------
// OUT[i, j] = b[j] + dinv[i]^2 * H[i, j]   (self-loop term + bias)
// ---------------------------------------------------------------------------
__global__ __launch_bounds__(256) void k_self_bias(const float* __restrict__ H,
                                                   const float* __restrict__ dinv,
                                                   const float* __restrict__ b,
                                                   float* __restrict__ OUT) {
    size_t idx = (size_t)blockIdx.x * 256 + threadIdx.x;
    if (idx < (size_t)N_NODES * D_HID) {
        int node = (int)(idx >> 8);
        int j    = (int)(idx & 255);
        float dv = dinv[node];
        OUT[idx] = b[j] + dv * dv * H[idx];
    }
}

// ---------------------------------------------------------------------------
// edge scatter: one wave per edge. OUT[c,:] += dinv[r]*w*dinv[c] * H[r,:]
// lane j handles features j, j+32, ..., j+224 -> coalesced 128B per step
// ---------------------------------------------------------------------------
__global__ __launch_bounds__(256) void k_edge_scatter(const int* __restrict__ row,
                                                      const int* __restrict__ col,
                                                      const float* __restrict__ w,
                                                      const float* __restrict__ dinv,
                                                      const float* __restrict__ H,
                                                      float* __restrict__ OUT) {
    const int e = blockIdx.x * 8 + (threadIdx.x >> 5);
    if (e >= N_EDGES) return;
    const int lane = threadIdx.x & 31;
    const int r = row[e];
    const int c = col[e];
    const float nrm = dinv[r] * w[e] * dinv[c];
    const float* hr = H + (size_t)r * D_HID;
    float* oc = OUT + (size_t)c * D_HID;
    #pragma unroll
    for (int i = 0; i < 8; ++i) {
        const int j = lane + (i << 5);
        atomicAddF32(oc + j, nrm * hr[j]);
    }
}

// ---------------------------------------------------------------------------
// column sums for mean pooling: gsum[j] += sum_i H[i, j]
// ---------------------------------------------------------------------------
__global__ __launch_bounds__(256) void k_colsum(const float* __restrict__ H,
                                                float* __restrict__ gsum) {
    const int t = threadIdx.x;
    float s = 0.0f;
    for (int i = blockIdx.x; i < N_NODES; i += gridDim.x)
        s += H[(size_t)i * D_HID + t];
    atomicAddF32(&gsum[t], s);
}

// ---------------------------------------------------------------------------
// dense head: g = gsum/N; z = relu(g @ Wd1 + bd1); softmax(z @ Wd2 + bd2)
// ---------------------------------------------------------------------------
__global__ __launch_bounds__(128) void k_head(const float* __restrict__ gsum,
                                              const float* __restrict__ Wd1,
                                              const float* __restrict__ bd1,
                                              const float* __restrict__ Wd2,
                                              const float* __restrict__ bd2,
                                              float* __restrict__ out) {
    __shared__ float z[D_DENSE];
    __shared__ float logits[N_CLASSES];
    const int j = threadIdx.x;                  // 0..127
    const float inv_n = 1.0f / (float)N_NODES;
    float acc = bd1[j];
    for (int k = 0; k < D_HID; ++k)
        acc += (gsum[k] * inv_n) * Wd1[k * D_DENSE + j];
    z[j] = fmaxf(acc, 0.0f);
    __syncthreads();
    if (j < N_CLASSES) {
        float l = bd2[j];
        for (int q = 0; q < D_DENSE; ++q)
            l += z[q] * Wd2[q * N_CLASSES + j];
        logits[j] = l;
    }
    __syncthreads();
    if (j == 0) {
        float m  = fmaxf(logits[0], logits[1]);
        float e0 = __expf(logits[0] - m);
        float e1 = __expf(logits[1] - m);
        float d  = e0 + e1;
        out[0] = e0 / d;
        out[1] = e1 / d;
    }
}

// ---------------------------------------------------------------------------
// launch
// ---------------------------------------------------------------------------
extern "C" void kernel_launch(void* const* d_in, const int* in_sizes, int n_in,
                              void* d_out, int out_size, void* d_ws, size_t ws_size,
                              hipStream_t stream) {
    const float* x   = (const float*)d_in[0];
    const int*   ei  = (const int*)  d_in[1];
    const float* ew  = (const float*)d_in[2];
    const float* W1  = (const float*)d_in[3];
    const float* b1  = (const float*)d_in[4];
    const float* W2  = (const float*)d_in[5];
    const float* b2  = (const float*)d_in[6];
    const float* Wd1 = (const float*)d_in[7];
    const float* bd1 = (const float*)d_in[8];
    const float* Wd2 = (const float*)d_in[9];
    const float* bd2 = (const float*)d_in[10];
    float* out = (float*)d_out;

    // workspace carve-up (floats): dinv | bufA (20.48MB) | bufB (20.48MB) | gsum
    float* ws   = (float*)d_ws;
    float* dinv = ws;
    float* bufA = ws + 32768;
    float* bufB = bufA + (size_t)N_NODES * D_HID;
    float* gsum = bufB + (size_t)N_NODES * D_HID;

    const int* row = ei;             // edge_index[0, :]
    const int* col = ei + N_EDGES;   // edge_index[1, :]

    // normalization
    k_init <<<(N_NODES + 255) / 256, 256, 0, stream>>>(dinv, gsum);
    k_deg  <<<(N_EDGES + 255) / 256, 256, 0, stream>>>(col, ew, dinv);
    k_rsqrt<<<(N_NODES + 255) / 256, 256, 0, stream>>>(dinv);

    const dim3 ggrid(N_NODES / 32);   // 625 blocks, each 32 rows x full 256 cols

    // layer 1: h1 = norm-agg(x @ W1) + b1
    k_gemm        <<<ggrid, 256, 0, stream>>>(x, W1, bufA, D_IN);
    k_self_bias   <<<N_NODES, 256, 0, stream>>>(bufA, dinv, b1, bufB);
    k_edge_scatter<<<(N_EDGES + 7) / 8, 256, 0, stream>>>(row, col, ew, dinv, bufA, bufB);

    // layer 2: h2 = norm-agg(h1 @ W2) + b2
    k_gemm        <<<ggrid, 256, 0, stream>>>(bufB, W2, bufA, D_HID);
    k_self_bias   <<<N_NODES, 256, 0, stream>>>(bufA, dinv, b2, bufB);
    k_edge_scatter<<<(N_EDGES + 7) / 8, 256, 0, stream>>>(row, col, ew, dinv, bufA, bufB);

    // mean pooling + MLP head + softmax
    k_colsum<<<256, 256, 0, stream>>>(bufB, gsum);
    k_head  <<<1, D_DENSE, 0, stream>>>(gsum, Wd1, bd1, Wd2, bd2, out);
}